// BCHConsolidator_25426206392459
// MI455X (gfx1250) — compile-verified
//
#include <hip/hip_runtime.h>
#include <math.h>

#define D_DIM   64
#define KVEC    2016            // D*(D-1)/2
#define LDSLD   65              // padded leading dim (bank-conflict avoidance)
#define ETA_C   0.05f
#define RADIUS_C 0.693f
#define NTHREADS 256

typedef __attribute__((ext_vector_type(2))) float v2f;
typedef __attribute__((ext_vector_type(8))) float v8f;

// offset of row i in triu_indices(D,1) row-major ordering
__device__ __forceinline__ int row_offset(int i) {
    return (i * (2 * D_DIM - i - 1)) >> 1;
}

// linear upper-tri index k -> (i, j), i < j
__device__ __forceinline__ void k_to_ij(int k, int& io, int& jo) {
    const float c = 2.0f * (float)D_DIM - 1.0f;            // 127
    int i = (int)floorf(0.5f * (c - sqrtf(c * c - 8.0f * (float)k)));
    if (i < 0) i = 0;
    if (i > D_DIM - 2) i = D_DIM - 2;
    while (i < D_DIM - 2 && row_offset(i + 1) <= k) ++i;   // fixup (<=2 steps)
    while (i > 0 && row_offset(i) > k) --i;
    io = i;
    jo = k - row_offset(i) + i + 1;
}

// Spectral norm (largest singular value) of a skew-symmetric 64x64 matrix in
// LDS via power iteration on A^T A = -A^2.  All 256 threads must enter.
__device__ float spectral_norm_skew(const float* __restrict__ M,
                                    float* pv, float* pw, float* pu, int tid) {
    if (tid < D_DIM) pv[tid] = 1.0f + 0.0625f * (float)(tid & 7);
    __syncthreads();
    for (int it = 0; it < 12; ++it) {
        if (tid < D_DIM) {                 // w = A v
            float s = 0.0f;
            #pragma unroll 8
            for (int j = 0; j < D_DIM; ++j) s = fmaf(M[tid * LDSLD + j], pv[j], s);
            pw[tid] = s;
        }
        __syncthreads();
        if (tid < D_DIM) {                 // u = A^T w = -A w
            float s = 0.0f;
            #pragma unroll 8
            for (int j = 0; j < D_DIM; ++j) s = fmaf(M[tid * LDSLD + j], pw[j], s);
            pu[tid] = -s;
        }
        __syncthreads();
        if (tid < D_DIM) pw[tid] = pu[tid] * pu[tid];
        __syncthreads();
        for (int off = 32; off > 0; off >>= 1) {
            if (tid < off) pw[tid] += pw[tid + off];
            __syncthreads();
        }
        float inv = rsqrtf(pw[0] + 1e-30f);
        __syncthreads();
        if (tid < D_DIM) pv[tid] = pu[tid] * inv;
        __syncthreads();
    }
    if (tid < D_DIM) {                     // sigma = ||A v||, ||v|| == 1
        float s = 0.0f;
        #pragma unroll 8
        for (int j = 0; j < D_DIM; ++j) s = fmaf(M[tid * LDSLD + j], pv[j], s);
        pw[tid] = s * s;
    }
    __syncthreads();
    for (int off = 32; off > 0; off >>= 1) {
        if (tid < off) pw[tid] += pw[tid + off];
        __syncthreads();
    }
    float sigma = sqrtf(pw[0]);
    __syncthreads();
    return sigma;
}

__global__ __launch_bounds__(NTHREADS)
void bch_consolidate(const float* __restrict__ fv, const int* __restrict__ ids,
                     const float* __restrict__ delta, float* __restrict__ out) {
    __shared__ float sA[D_DIM * LDSLD];    // A_old (skew)
    __shared__ float sB[D_DIM * LDSLD];    // dA -> dAs (scaled skew delta)
    __shared__ float sC[D_DIM * LDSLD];    // staging raw delta, then A_new
    __shared__ float pv[D_DIM], pw[D_DIM], pu[D_DIM];
    __shared__ float s_scale;

    const int tid = threadIdx.x;
    const int b   = blockIdx.x;
    const int row = ids[b];
    const float* fvrow = fv    + (long long)row * KVEC;
    const float* drow  = delta + (long long)b * (D_DIM * D_DIM);

    // ---- Phase 1: gather skew A_old into sA, raw delta into sC ------------
    for (int k = tid; k < KVEC; k += NTHREADS) {
        int i, j; k_to_ij(k, i, j);
        float val = fvrow[k];
        sA[i * LDSLD + j] =  val;
        sA[j * LDSLD + i] = -val;
    }
    if (tid < D_DIM) sA[tid * LDSLD + tid] = 0.0f;
    for (int e = tid; e < (D_DIM * D_DIM) / 4; e += NTHREADS) {
        const float4 q = ((const float4*)drow)[e];     // global_load_b128
        int base = e * 4;
        int i = base >> 6, j = base & 63;
        sC[i * LDSLD + j + 0] = q.x;
        sC[i * LDSLD + j + 1] = q.y;
        sC[i * LDSLD + j + 2] = q.z;
        sC[i * LDSLD + j + 3] = q.w;
    }
    __syncthreads();

    // ---- Phase 2: dA = 0.5*(X - X^T) into sB ------------------------------
    for (int e = tid; e < D_DIM * D_DIM; e += NTHREADS) {
        int i = e >> 6, j = e & 63;
        sB[i * LDSLD + j] = 0.5f * (sC[i * LDSLD + j] - sC[j * LDSLD + i]);
    }
    __syncthreads();

    // ---- Phase 3: trust-region scale from spectral norms ------------------
    float norm_old = spectral_norm_skew(sA, pv, pw, pu, tid);
    float norm_dA  = spectral_norm_skew(sB, pv, pw, pu, tid);
    if (tid == 0) {
        float norm_delta = ETA_C * norm_dA;
        float avail = fmaxf(RADIUS_C - norm_old, 1e-8f);
        s_scale = fminf(avail / (norm_delta + 1e-8f), 1.0f);
    }
    __syncthreads();
    float scl = s_scale;
    for (int e = tid; e < D_DIM * D_DIM; e += NTHREADS) {
        int i = e >> 6, j = e & 63;
        sB[i * LDSLD + j] *= scl;                       // dAs
    }
    __syncthreads();

    // ---- Phase 4: bracket via V_WMMA_F32_16X16X4_F32, build A_new in sC ---
    // 16 output tiles of 16x16; 8 waves -> 2 tiles each; wave-uniform control
    // flow so EXEC is all-ones at every WMMA (ISA 7.12 requirement).
    const int lane = tid & 31;
    const int wave = tid >> 5;
    const int half = lane >> 4;     // 0: lanes 0-15, 1: lanes 16-31
    const int l16  = lane & 15;
    for (int t = wave; t < 16; t += 8) {
        const int tm = (t >> 2) * 16;
        const int tn = (t & 3)  * 16;
        v8f acc1 = {};              // (A_old @ dAs) tile
        v8f acc2 = {};              // (dAs @ A_old) tile
        for (int k0 = 0; k0 < D_DIM; k0 += 4) {
            const int ka = k0 + 2 * half;
            v2f a1, b1, a2, b2;
            // A frag: lane(l16) = row m; VGPR0/1 = K = ka, ka+1
            a1.x = sA[(tm + l16) * LDSLD + ka];
            a1.y = sA[(tm + l16) * LDSLD + ka + 1];
            // B frag: lane(l16) = col n; VGPR0/1 = K = ka, ka+1
            b1.x = sB[(ka)     * LDSLD + tn + l16];
            b1.y = sB[(ka + 1) * LDSLD + tn + l16];
            acc1 = __builtin_amdgcn_wmma_f32_16x16x4_f32(
                       false, a1, false, b1, (short)0, acc1, false, false);
            a2.x = sB[(tm + l16) * LDSLD + ka];
            a2.y = sB[(tm + l16) * LDSLD + ka + 1];
            b2.x = sA[(ka)     * LDSLD + tn + l16];
            b2.y = sA[(ka + 1) * LDSLD + tn + l16];
            acc2 = __builtin_amdgcn_wmma_f32_16x16x4_f32(
                       false, a2, false, b2, (short)0, acc2, false, false);
        }
        // C/D layout: VGPR r -> row tm + r + 8*half, col tn + l16
        #pragma unroll
        for (int r = 0; r < 8; ++r) {
            int m = tm + r + 8 * half;
            int n = tn + l16;
            float bracket = acc1[r] - acc2[r];
            sC[m * LDSLD + n] = sA[m * LDSLD + n]
                              + ETA_C * sB[m * LDSLD + n]
                              + 0.5f * ETA_C * bracket;
        }
    }
    __syncthreads();

    // ---- Phase 5: re-project to so(D) -------------------------------------
    for (int k = tid; k < KVEC; k += NTHREADS) {
        int i, j; k_to_ij(k, i, j);
        float vv = 0.5f * (sC[i * LDSLD + j] - sC[j * LDSLD + i]);
        sC[i * LDSLD + j] =  vv;
        sC[j * LDSLD + i] = -vv;
    }
    if (tid < D_DIM) sC[tid * LDSLD + tid] = 0.0f;
    __syncthreads();

    // ---- Phase 6: clamp to BCH radius, scatter row ------------------------
    float fnorm  = spectral_norm_skew(sC, pv, pw, pu, tid);
    float scale2 = fminf(RADIUS_C / (fnorm + 1e-8f), 1.0f);
    float* outrow = out + (long long)row * KVEC;
    for (int k = tid; k < KVEC; k += NTHREADS) {
        int i, j; k_to_ij(k, i, j);
        outrow[k] = sC[i * LDSLD + j] * scale2;
    }
}

extern "C" void kernel_launch(void* const* d_in, const int* in_sizes, int n_in,
                              void* d_out, int out_size, void* d_ws, size_t ws_size,
                              hipStream_t stream) {
    const float* fv    = (const float*)d_in[0];   // (N_USERS, K) f32
    const int*   ids   = (const int*)  d_in[1];   // (B,) i32
    const float* delta = (const float*)d_in[2];   // (B, D, D) f32
    float* out = (float*)d_out;                   // (N_USERS, K) f32

    // Bulk copy of the fiber store (HBM-bound part, ~403 MB each way).
    hipMemcpyAsync(d_out, d_in[0], (size_t)out_size * sizeof(float),
                   hipMemcpyDeviceToDevice, stream);

    const int B = in_sizes[1];                    // 8192 unique rows
    bch_consolidate<<<B, NTHREADS, 0, stream>>>(fv, ids, delta, out);
}